// GATConv_27779848471359
// MI455X (gfx1250) — compile-verified
//
#include <hip/hip_runtime.h>
#include <hip/hip_bf16.h>

// GATConv for MI455X (gfx1250, wave32).
// Phase 1: feat = x @ W  via V_WMMA_F32_16X16X4_F32 (fp32 matrix pipe, exact vs f32 ref)
// Phase 2: el/er per (node, head)
// Phase 3: per-node softmax over 16 neighbors + weighted gather (wave32 shuffles)

typedef __attribute__((ext_vector_type(2))) float v2f;
typedef __attribute__((ext_vector_type(8))) float v8f;

#define GAT_IN_CH   128
#define GAT_HC      64      // HEADS * OUT_CH
#define GAT_HEADS   4
#define GAT_OUT_CH  16
#define NEG_SLOPE   0.2f

// ---------------------------------------------------------------------------
// Kernel 1: feat[n, hc] = sum_k x[n,k] * W[k,hc]
// Block = 256 threads = 8 waves; block tile = 32 rows x 64 cols.
// Wave w: row-subtile rt = w>>2 (0/1), col tile = (w&3)*16  (== one head).
// K loop: 32 steps of V_WMMA_F32_16X16X4_F32.
// Requires n_nodes % 32 == 0 (65536 here).
// ---------------------------------------------------------------------------
__global__ __launch_bounds__(256) void gat_gemm_feat(
    const float* __restrict__ x, const float* __restrict__ W,
    float* __restrict__ feat)
{
    __shared__ __align__(16) float sW[GAT_IN_CH * GAT_HC];  // 32 KB
    __shared__ __align__(16) float sX[32 * GAT_IN_CH];      // 16 KB

    const int tid = threadIdx.x;

    // Stage W (8192 floats = 2048 float4) and 32 rows of x (4096 floats = 1024 float4).
    const float4* Wv = (const float4*)W;
    float4* sWv = (float4*)sW;
#pragma unroll
    for (int i = 0; i < 8; ++i) sWv[tid + i * 256] = Wv[tid + i * 256];

    const int rowbase_blk = blockIdx.x * 32;
    const float4* Xv = (const float4*)(x + (size_t)rowbase_blk * GAT_IN_CH);
    float4* sXv = (float4*)sX;
#pragma unroll
    for (int i = 0; i < 4; ++i) sXv[tid + i * 256] = Xv[tid + i * 256];

    __syncthreads();

    const int wave    = tid >> 5;
    const int lane    = tid & 31;
    const int half    = lane >> 4;   // 0: lanes 0-15, 1: lanes 16-31
    const int m       = lane & 15;
    const int rt      = wave >> 2;          // 0..1
    const int colbase = (wave & 3) * 16;    // head * OUT_CH

    const float* As = sX + (rt * 16 + m) * GAT_IN_CH;

    v8f acc = {};
#pragma unroll
    for (int ks = 0; ks < 32; ++ks) {
        const int k0 = ks * 4 + half * 2;
        // A 16x4 f32 frag (ISA 7.12.2): lanes 0-15 -> K = {4ks, 4ks+1}; lanes 16-31 -> {4ks+2, 4ks+3}
        v2f a;
        a.x = As[k0];
        a.y = As[k0 + 1];
        // B 4x16 f32 frag: VGPR v, lane l -> K = v + 2*(l>=16), N = l&15
        v2f b;
        b.x = sW[(k0    ) * GAT_HC + colbase + m];
        b.y = sW[(k0 + 1) * GAT_HC + colbase + m];
        acc = __builtin_amdgcn_wmma_f32_16x16x4_f32(
                  /*neg_a=*/false, a, /*neg_b=*/false, b,
                  /*c_mod=*/(short)0, acc, /*reuse_a=*/false, /*reuse_b=*/false);
    }

    // C/D layout: VGPR v -> M = v + 8*half, N = lane&15
    const int row0 = rowbase_blk + rt * 16 + half * 8;
    float* outp = feat + (size_t)row0 * GAT_HC + colbase + m;
#pragma unroll
    for (int v = 0; v < 8; ++v) outp[(size_t)v * GAT_HC] = acc[v];
}

// ---------------------------------------------------------------------------
// Kernel 2: el[n,h] = feat[n,h,:] . attn_l[h,:],  er likewise.
// One thread per (node, head).
// ---------------------------------------------------------------------------
__global__ __launch_bounds__(256) void gat_eler(
    const float* __restrict__ feat,
    const float* __restrict__ attn_l, const float* __restrict__ attn_r,
    float* __restrict__ el, float* __restrict__ er, int n_nodes)
{
    int t = blockIdx.x * 256 + threadIdx.x;
    if (t >= n_nodes * GAT_HEADS) return;
    int n = t >> 2;
    int h = t & 3;
    const float4* f4 = (const float4*)(feat + (size_t)n * GAT_HC + h * GAT_OUT_CH);
    const float4* al = (const float4*)(attn_l + h * GAT_OUT_CH);
    const float4* ar = (const float4*)(attn_r + h * GAT_OUT_CH);
    float sl = 0.f, sr = 0.f;
#pragma unroll
    for (int i = 0; i < 4; ++i) {
        float4 f = f4[i], a = al[i], b = ar[i];
        sl += f.x * a.x + f.y * a.y + f.z * a.z + f.w * a.w;
        sr += f.x * b.x + f.y * b.y + f.z * b.z + f.w * b.w;
    }
    el[t] = sl;
    er[t] = sr;
}

// ---------------------------------------------------------------------------
// Kernel 3: per-node softmax over neighbors + weighted feature aggregation.
// One wave32 per node (8 nodes / 256-thread block).
// Lanes 0..deg-1 own edges (deg <= 32; fixed 16 in this problem).
// Accumulation: lane = (c, half); half 0 sums even edges, half 1 odd edges,
// combined with shfl_xor(16). Channels c = 0..15 per head.
// ---------------------------------------------------------------------------
__global__ __launch_bounds__(256) void gat_edge(
    const int* __restrict__ row_ptr, const int* __restrict__ col_ind,
    const float* __restrict__ feat,
    const float* __restrict__ el, const float* __restrict__ er,
    const float* __restrict__ bias,
    float* __restrict__ out, int n_nodes)
{
    int node = blockIdx.x * 8 + (threadIdx.x >> 5);
    if (node >= n_nodes) return;          // whole-wave uniform
    const int lane = threadIdx.x & 31;
    const int start = row_ptr[node];
    int deg = row_ptr[node + 1] - start;
    if (deg > 32) deg = 32;               // graph here has fixed degree 16

    const bool act = lane < deg;
    int j = 0;
    if (act) j = col_ind[start + lane];

    const int c    = lane & 15;
    const int half = lane >> 4;

#pragma unroll
    for (int h = 0; h < GAT_HEADS; ++h) {
        // logits + leaky relu
        float w = act ? (el[node * GAT_HEADS + h] + er[j * GAT_HEADS + h]) : -3.0e38f;
        w = (w >= 0.f) ? w : NEG_SLOPE * w;
        // segment max over wave
        float mx = w;
#pragma unroll
        for (int off = 16; off >= 1; off >>= 1) mx = fmaxf(mx, __shfl_xor(mx, off));
        // exp + segment sum
        float ew = act ? expf(w - mx) : 0.f;
        float s = ew;
#pragma unroll
        for (int off = 16; off >= 1; off >>= 1) s += __shfl_xor(s, off);
        const float inv = 1.f / s;

        // weighted aggregation of feat[j, h, c]
        float accv = 0.f;
        for (int e = half; e < deg; e += 2) {
            float a  = __shfl(ew, e) * inv;
            int   jj = __shfl(j, e);
            accv += a * feat[(size_t)jj * GAT_HC + h * GAT_OUT_CH + c];
        }
        accv += __shfl_xor(accv, 16);
        if (lane < 16)
            out[(size_t)node * GAT_HC + h * GAT_OUT_CH + c] = accv + bias[h * GAT_OUT_CH + c];
    }
}

// ---------------------------------------------------------------------------
// Host launcher.
// Inputs (setup_inputs order):
//  0 csr_row_ptr (int), 1 csr_col_ind (int), 2 sample_count (unused),
//  3 x_neighboor (f32), 4 x_target (unused by reference), 5 W (f32),
//  6 attn_l (f32), 7 attn_r (f32), 8 bias (f32)
// Workspace layout: feat (N*64 f32) | el (N*4 f32) | er (N*4 f32)  (~18 MB)
// ---------------------------------------------------------------------------
extern "C" void kernel_launch(void* const* d_in, const int* in_sizes, int n_in,
                              void* d_out, int out_size, void* d_ws, size_t ws_size,
                              hipStream_t stream) {
    const int*   row_ptr = (const int*)d_in[0];
    const int*   col_ind = (const int*)d_in[1];
    const float* x       = (const float*)d_in[3];
    const float* W       = (const float*)d_in[5];
    const float* attn_l  = (const float*)d_in[6];
    const float* attn_r  = (const float*)d_in[7];
    const float* bias    = (const float*)d_in[8];
    float*       out     = (float*)d_out;

    const int n_nodes = in_sizes[0] - 1;   // 65536

    float* feat = (float*)d_ws;
    float* el   = feat + (size_t)n_nodes * GAT_HC;
    float* er   = el + (size_t)n_nodes * GAT_HEADS;

    // Phase 1: WMMA GEMM (n_nodes divisible by 32)
    gat_gemm_feat<<<n_nodes / 32, 256, 0, stream>>>(x, W, feat);

    // Phase 2: attention projections
    gat_eler<<<(n_nodes * GAT_HEADS + 255) / 256, 256, 0, stream>>>(
        feat, attn_l, attn_r, el, er, n_nodes);

    // Phase 3: softmax + aggregate
    gat_edge<<<(n_nodes + 7) / 8, 256, 0, stream>>>(
        row_ptr, col_ind, feat, el, er, bias, out, n_nodes);
}